// CornerGNN_4784593567781
// MI455X (gfx1250) — compile-verified
//
#include <hip/hip_runtime.h>
#include <hip/hip_bf16.h>

typedef float v2f __attribute__((ext_vector_type(2)));
typedef float v8f __attribute__((ext_vector_type(8)));

#define BN_EPS 1e-5f

static __host__ int cdiv(int a, int b) { return (a + b - 1) / b; }

// ---------------- zero-fill ----------------
__global__ void k_zero_f32(float* __restrict__ p, int n) {
  int i = blockIdx.x * blockDim.x + threadIdx.x;
  if (i < n) p[i] = 0.0f;
}

// ---------------- pad rows from Kin to Kout (zero fill) ----------------
__global__ void k_pad(const float* __restrict__ src, float* __restrict__ dst,
                      int rows, int Kin, int Kout) {
  int t = blockIdx.x * blockDim.x + threadIdx.x;
  if (t >= rows * Kout) return;
  int r = t / Kout;                  // Kout power of two
  int c = t - r * Kout;
  dst[t] = (c < Kin) ? src[(size_t)r * Kin + c] : 0.0f;
}

// ---------------- degree / dinv ----------------
__global__ void k_deg(const int* __restrict__ dst, float* __restrict__ deg, int E) {
  int e = blockIdx.x * blockDim.x + threadIdx.x;
  if (e < E) atomicAdd(&deg[dst[e]], 1.0f);
}
__global__ void k_dinv(float* __restrict__ deg, int n) {
  int i = blockIdx.x * blockDim.x + threadIdx.x;
  if (i < n) deg[i] = rsqrtf(deg[i] + 1.0f);  // +1 self-loop
}

// ------- fp32 WMMA GEMM: C[M,NT*16] = A[M,K] @ W[NT*16,K]^T (+bias)(+relu) -------
// K must be a multiple of 4. One wave per 16-row strip; A fragments live in
// registers across all NT column tiles; W staged in LDS (row-padded, conflict-free).
// B fragments for all NT tiles are batch-loaded into distinct regs before the
// WMMA chain so one dscnt wait covers NT independent ds_loads.
template <int NT>
__global__ __launch_bounds__(256) void k_gemm_wmma(
    const float* __restrict__ A, const float* __restrict__ W,
    const float* __restrict__ bias, float* __restrict__ C,
    int M, int K, int relu)
{
  extern __shared__ float sW[];            // [NT*16][K+2]
  const int Ncol = NT * 16;
  const int Kp = K + 2;

  // cooperative stage of W into LDS
  for (int idx = threadIdx.x; idx < Ncol * K; idx += 256) {
    int r = idx / K;                       // K power of two here
    int c = idx - r * K;
    sW[r * Kp + c] = W[idx];
  }
  __syncthreads();

  const int lane = threadIdx.x & 31;
  const int wave = threadIdx.x >> 5;
  const int l16  = lane & 15;
  const int half = lane >> 4;              // 0: lanes 0-15, 1: lanes 16-31
  const int row0 = (blockIdx.x * 8 + wave) * 16;
  if (row0 >= M) return;                   // wave-uniform exit (after barrier)

  int ar = row0 + l16;
  if (ar >= M) ar = M - 1;                 // branch-free tail clamp; stores are guarded
  const float* Arow = A + (size_t)ar * K + 2 * half;
  const float* sB   = &sW[l16 * Kp + 2 * half];

  v8f acc[NT];
#pragma unroll
  for (int t = 0; t < NT; ++t) acc[t] = (v8f){};

  for (int k0 = 0; k0 < K; k0 += 4) {
    const v2f a = *(const v2f*)(Arow + k0);        // global_load_b64
    v2f b[NT];
#pragma unroll
    for (int t = 0; t < NT; ++t)                   // NT independent ds_loads
      b[t] = *(const v2f*)(sB + (size_t)t * 16 * Kp + k0);
#pragma unroll
    for (int t = 0; t < NT; ++t)
      acc[t] = __builtin_amdgcn_wmma_f32_16x16x4_f32(
          /*neg_a=*/false, a, /*neg_b=*/false, b[t],
          /*c_mod=*/(short)0, acc[t], /*reuse_a=*/false, /*reuse_b=*/false);
  }

#pragma unroll
  for (int t = 0; t < NT; ++t) {
    const int col = t * 16 + l16;
    const float bv = bias ? bias[col] : 0.0f;
#pragma unroll
    for (int v = 0; v < 8; ++v) {
      const int row = row0 + v + 8 * half;  // C/D layout: VGPR v -> M=v (lo) / v+8 (hi)
      if (row < M) {
        float o = acc[t][v] + bv;
        if (relu) o = fmaxf(o, 0.0f);
        C[(size_t)row * Ncol + col] = o;
      }
    }
  }
}

// ---------------- edge gather * norm + scatter-add (float4 / thread) ----------------
__global__ void k_scatter(const int* __restrict__ src, const int* __restrict__ dst,
                          const float* __restrict__ dinv, const float* __restrict__ h,
                          float* __restrict__ agg, int E, int D, int lgv)
{
  int t = blockIdx.x * blockDim.x + threadIdx.x;
  int e = t >> lgv;                        // lgv = log2(D/4)
  if (e >= E) return;
  int j = (t & ((1 << lgv) - 1)) << 2;
  int s = src[e], d = dst[e];
  float norm = dinv[s] * dinv[d];
  float4 hv = *(const float4*)(h + (size_t)s * D + j);
  float* ap = agg + (size_t)d * D + j;
  atomicAdd(ap + 0, hv.x * norm);
  atomicAdd(ap + 1, hv.y * norm);
  atomicAdd(ap + 2, hv.z * norm);
  atomicAdd(ap + 3, hv.w * norm);
}

// ---------------- self-loop + bias + BN + ReLU ----------------
__global__ void k_epilogue(const float* __restrict__ agg, const float* __restrict__ h,
                           const float* __restrict__ dinv,
                           const float* __restrict__ b, const float* __restrict__ g,
                           const float* __restrict__ beta, const float* __restrict__ m,
                           const float* __restrict__ var,
                           float* __restrict__ out, int n, int D)
{
  int t = blockIdx.x * blockDim.x + threadIdx.x;
  if (t >= n * D) return;
  int i = t / D;
  int c = t - i * D;
  float di = dinv[i];
  float val = agg[t] + h[t] * di * di + b[c];
  val = (val - m[c]) * rsqrtf(var[c] + BN_EPS) * g[c] + beta[c];
  out[t] = fmaxf(val, 0.0f);
}

// ---------------- pooling ----------------
__global__ void k_cnt(const int* __restrict__ batch, float* __restrict__ cnt, int n) {
  int i = blockIdx.x * blockDim.x + threadIdx.x;
  if (i < n) atomicAdd(&cnt[batch[i]], 1.0f);
}
__global__ void k_pool(const float* __restrict__ h, const int* __restrict__ batch,
                       float* __restrict__ sum, int* __restrict__ mx, int n, int D)
{
  int t = blockIdx.x * blockDim.x + threadIdx.x;
  if (t >= n * D) return;
  int i = t / D;
  int c = t - i * D;
  int g = batch[i];
  float val = h[t];
  atomicAdd(&sum[(size_t)g * D + c], val);
  atomicMax(&mx[(size_t)g * D + c], __float_as_int(val));  // val >= 0 post-ReLU
}
__global__ void k_buildz(const float* __restrict__ sum, const int* __restrict__ mx,
                         const float* __restrict__ cnt, float* __restrict__ z,
                         int G, int D)
{
  int t = blockIdx.x * blockDim.x + threadIdx.x;
  if (t >= G * 2 * D) return;
  int g = t / (2 * D);
  int c = t - g * 2 * D;
  if (c < D) z[t] = sum[(size_t)g * D + c] / fmaxf(cnt[g], 1.0f);
  else       z[t] = __int_as_float(mx[(size_t)g * D + (c - D)]);
}

// ---------------- final fc3: [G,32] -> [G,1] ----------------
__global__ void k_fc3(const float* __restrict__ z2, const float* __restrict__ w,
                      const float* __restrict__ b, float* __restrict__ out, int G, int K)
{
  int g = blockIdx.x * blockDim.x + threadIdx.x;
  if (g >= G) return;
  float acc = b[0];
  for (int k = 0; k < K; ++k) acc += z2[(size_t)g * K + k] * w[k];
  out[g] = acc;
}

extern "C" void kernel_launch(void* const* d_in, const int* in_sizes, int n_in,
                              void* d_out, int out_size, void* d_ws, size_t ws_size,
                              hipStream_t stream) {
  const float* x     = (const float*)d_in[0];
  const int*   ei    = (const int*)d_in[1];
  const int*   batch = (const int*)d_in[2];
  const float* W1 = (const float*)d_in[3],  *b1 = (const float*)d_in[4];
  const float* g1 = (const float*)d_in[5],  *be1 = (const float*)d_in[6];
  const float* m1 = (const float*)d_in[7],  *v1 = (const float*)d_in[8];
  const float* W2 = (const float*)d_in[9],  *b2 = (const float*)d_in[10];
  const float* g2 = (const float*)d_in[11], *be2 = (const float*)d_in[12];
  const float* m2 = (const float*)d_in[13], *v2 = (const float*)d_in[14];
  const float* W3 = (const float*)d_in[15], *b3 = (const float*)d_in[16];
  const float* g3 = (const float*)d_in[17], *be3 = (const float*)d_in[18];
  const float* m3 = (const float*)d_in[19], *v3 = (const float*)d_in[20];
  const float* fc1W = (const float*)d_in[21], *fc1b = (const float*)d_in[22];
  const float* fc2W = (const float*)d_in[23], *fc2b = (const float*)d_in[24];
  const float* fc3W = (const float*)d_in[25], *fc3b = (const float*)d_in[26];

  const int N = in_sizes[0] / 14;
  const int E = in_sizes[1] / 2;
  const int G = out_size;          // [G,1] logits
  const int* src = ei;
  const int* dst = ei + E;

  // workspace layout (floats)
  float* ws   = (float*)d_ws;
  float* dinv = ws;                         // N
  float* bufH = dinv + N;                   // N*128
  float* bufA = bufH + (size_t)N * 128;     // N*128 (scatter target)
  float* bufB = bufA + (size_t)N * 128;     // N*128 (activations)
  float* xp   = bufB + (size_t)N * 128;     // N*16  (x padded K 14->16)
  float* W1p  = xp + (size_t)N * 16;        // 64*16 (W1 padded)
  float* cnt  = W1p + 64 * 16;              // G
  float* psum = cnt + G;                    // G*64
  int*   pmax = (int*)(psum + (size_t)G * 64);   // G*64
  float* z    = (float*)(pmax + (size_t)G * 64); // G*128
  float* z1   = z + (size_t)G * 128;        // G*64
  float* z2   = z1 + (size_t)G * 64;        // G*32

  const int B = 256;

  // ---- pad x and W1 to K=16 ----
  k_pad<<<cdiv(N * 16, B), B, 0, stream>>>(x, xp, N, 14, 16);
  k_pad<<<cdiv(64 * 16, B), B, 0, stream>>>(W1, W1p, 64, 14, 16);

  // ---- degree -> dinv ----
  k_zero_f32<<<cdiv(N, B), B, 0, stream>>>(dinv, N);
  k_deg<<<cdiv(E, B), B, 0, stream>>>(dst, dinv, E);
  k_dinv<<<cdiv(N, B), B, 0, stream>>>(dinv, N);

  // ================= GCN layer 1: 16(padded) -> 64 =================
  {
    size_t lds = 64 * (16 + 2) * sizeof(float);
    k_gemm_wmma<4><<<cdiv(N, 128), B, lds, stream>>>(xp, W1p, nullptr, bufH, N, 16, 0);
    k_zero_f32<<<cdiv(N * 64, B), B, 0, stream>>>(bufA, N * 64);
    k_scatter<<<cdiv(E << 4, B), B, 0, stream>>>(src, dst, dinv, bufH, bufA, E, 64, 4);
    k_epilogue<<<cdiv(N * 64, B), B, 0, stream>>>(bufA, bufH, dinv, b1, g1, be1, m1, v1,
                                                  bufB, N, 64);
  }
  // ================= GCN layer 2: 64 -> 128 =================
  {
    size_t lds = 128 * (64 + 2) * sizeof(float);
    k_gemm_wmma<8><<<cdiv(N, 128), B, lds, stream>>>(bufB, W2, nullptr, bufH, N, 64, 0);
    k_zero_f32<<<cdiv(N * 128, B), B, 0, stream>>>(bufA, N * 128);
    k_scatter<<<cdiv(E << 5, B), B, 0, stream>>>(src, dst, dinv, bufH, bufA, E, 128, 5);
    k_epilogue<<<cdiv(N * 128, B), B, 0, stream>>>(bufA, bufH, dinv, b2, g2, be2, m2, v2,
                                                   bufB, N, 128);
  }
  // ================= GCN layer 3: 128 -> 64 =================
  {
    size_t lds = 64 * (128 + 2) * sizeof(float);
    k_gemm_wmma<4><<<cdiv(N, 128), B, lds, stream>>>(bufB, W3, nullptr, bufH, N, 128, 0);
    k_zero_f32<<<cdiv(N * 64, B), B, 0, stream>>>(bufA, N * 64);
    k_scatter<<<cdiv(E << 4, B), B, 0, stream>>>(src, dst, dinv, bufH, bufA, E, 64, 4);
    k_epilogue<<<cdiv(N * 64, B), B, 0, stream>>>(bufA, bufH, dinv, b3, g3, be3, m3, v3,
                                                  bufB, N, 64);
  }

  // ---- pooling: mean + max per graph over 64 channels ----
  k_zero_f32<<<cdiv(G, B), B, 0, stream>>>(cnt, G);
  k_zero_f32<<<cdiv(G * 64, B), B, 0, stream>>>(psum, G * 64);
  k_zero_f32<<<cdiv(G * 64, B), B, 0, stream>>>((float*)pmax, G * 64);  // 0 bits == 0.0f
  k_cnt<<<cdiv(N, B), B, 0, stream>>>(batch, cnt, N);
  k_pool<<<cdiv(N * 64, B), B, 0, stream>>>(bufB, batch, psum, pmax, N, 64);
  k_buildz<<<cdiv(G * 128, B), B, 0, stream>>>(psum, pmax, cnt, z, G, 64);

  // ---- MLP head ----
  {
    size_t lds = 64 * (128 + 2) * sizeof(float);
    k_gemm_wmma<4><<<cdiv(G, 128), B, lds, stream>>>(z, fc1W, fc1b, z1, G, 128, 1);
  }
  {
    size_t lds = 32 * (64 + 2) * sizeof(float);
    k_gemm_wmma<2><<<cdiv(G, 128), B, lds, stream>>>(z1, fc2W, fc2b, z2, G, 64, 1);
  }
  k_fc3<<<cdiv(G, B), B, 0, stream>>>(z2, fc3W, fc3b, (float*)d_out, G, 32);
}